// TransitionDown_51694226375250
// MI455X (gfx1250) — compile-verified
//
#include <hip/hip_runtime.h>
#include <stdint.h>

typedef float v2f __attribute__((ext_vector_type(2)));
typedef float v8f __attribute__((ext_vector_type(8)));

namespace {
constexpr int kB = 4, kN = 4096, kK = 16, kS = 1024;
constexpr int kCin = 67, kCinP = 68, kCout = 128;
constexpr int kM = kB * kN * kK;              // 262144 sample rows
constexpr int kRowsPerWG = 64;
constexpr int kNWG = kM / kRowsPerWG;         // 4096 workgroups per GEMM
// workspace byte offsets
constexpr size_t OFF_CENT = 0;                               // B*S ints
constexpr size_t OFF_KNN  = OFF_CENT + (size_t)kB * kS * 4;  // M ints (flat == row index)
constexpr size_t OFF_S1   = OFF_KNN + (size_t)kM * 4;
constexpr size_t OFF_T1   = OFF_S1 + 512;
constexpr size_t OFF_S2   = OFF_T1 + 512;
constexpr size_t OFF_T2   = OFF_S2 + 512;
constexpr size_t OFF_PART = ((OFF_T2 + 512 + 255) / 256) * 256;             // kNWG*256 f32
constexpr size_t OFF_H    = ((OFF_PART + (size_t)kNWG * 256 * 4 + 255) / 256) * 256; // M*128 f32
}

// ---------------------------------------------------------------- FPS ----
__device__ __forceinline__ void argmax_red(float& v, int& i) {
#pragma unroll
  for (int off = 16; off > 0; off >>= 1) {
    float ov = __shfl_xor(v, off, 32);
    int   oi = __shfl_xor(i, off, 32);
    if (ov > v || (ov == v && oi < i)) { v = ov; i = oi; }
  }
}

__global__ void fps_kernel(const float* __restrict__ pos, int* __restrict__ cent) {
  __shared__ float dists[kN];
  __shared__ unsigned char flg[kN];
  __shared__ float wv[32];
  __shared__ int   wi[32];
  __shared__ int   cnts[1024];
  __shared__ float s_pl[3];
  __shared__ int   s_last;

  const int b = blockIdx.x, tid = threadIdx.x;
  const int lane = tid & 31, wav = tid >> 5;
  const float* bp = pos + (size_t)b * kN * 3;

  for (int p = tid; p < kN; p += 1024) { dists[p] = 1e10f; flg[p] = 0; }
  if (tid == 0) { s_last = 0; flg[0] = 1; }
  __syncthreads();

  for (int it = 1; it < kS; ++it) {
    if (tid < 3) s_pl[tid] = bp[(size_t)s_last * 3 + tid];
    __syncthreads();
    const float px = s_pl[0], py = s_pl[1], pz = s_pl[2];
    float bv = -1.0f; int bi = 1 << 30;
#pragma unroll
    for (int rep = 0; rep < kN / 1024; ++rep) {
      const int p = tid + rep * 1024;
      const float dx = bp[p * 3 + 0] - px;
      const float dy = bp[p * 3 + 1] - py;
      const float dz = bp[p * 3 + 2] - pz;
      const float d = dx * dx + dy * dy + dz * dz;
      const float nd = fminf(dists[p], d);
      dists[p] = nd;
      if (nd > bv || (nd == bv && p < bi)) { bv = nd; bi = p; }
    }
    argmax_red(bv, bi);
    if (lane == 0) { wv[wav] = bv; wi[wav] = bi; }
    __syncthreads();
    if (wav == 0) {
      float v = wv[lane]; int i = wi[lane];
      argmax_red(v, i);
      if (lane == 0) { s_last = i; flg[i] = 1; }
    }
    __syncthreads();
  }
  // ascending compaction of selected flags (matches jnp.sort of FPS indices)
  int c = 0;
#pragma unroll
  for (int e = 0; e < 4; ++e) c += flg[tid * 4 + e];
  cnts[tid] = c;
  __syncthreads();
  if (tid == 0) {
    int acc = 0;
    for (int t = 0; t < 1024; ++t) { int tv = cnts[t]; cnts[t] = acc; acc += tv; }
  }
  __syncthreads();
  int o = cnts[tid];
#pragma unroll
  for (int e = 0; e < 4; ++e) {
    const int p = tid * 4 + e;
    if (flg[p]) cent[(size_t)b * kS + (o++)] = p;
  }
}

// ---------------------------------------------------------------- KNN ----
__global__ void knn_kernel(const float* __restrict__ pos, int* __restrict__ nbr) {
  __shared__ float cp[1024 * 3];
  const int tid = threadIdx.x;
  const int b = blockIdx.x >> 4;                 // 16 blocks per cloud
  const int q = ((blockIdx.x & 15) << 8) + tid;  // query point in cloud
  const size_t gq = (size_t)b * kN + q;

  const float qx = pos[gq * 3 + 0], qy = pos[gq * 3 + 1], qz = pos[gq * 3 + 2];
  float qd[kK]; int qi[kK];
#pragma unroll
  for (int r = 0; r < kK; ++r) { qd[r] = 3e38f; qi[r] = -1; }

  for (int tile = 0; tile < 4; ++tile) {
    __syncthreads();
    for (int idx = tid; idx < 3072; idx += 256)
      cp[idx] = pos[((size_t)b * kN + tile * 1024) * 3 + idx];
    __syncthreads();
    for (int j = 0; j < 1024; ++j) {
      const float dx = cp[j * 3 + 0] - qx;
      const float dy = cp[j * 3 + 1] - qy;
      const float dz = cp[j * 3 + 2] - qz;
      float d = dx * dx + dy * dy + dz * dz;
      if (d < qd[kK - 1]) {               // strict < keeps earlier index on ties
        qd[kK - 1] = d; qi[kK - 1] = tile * 1024 + j;
#pragma unroll
        for (int r = kK - 1; r >= 1; --r) {
          if (qd[r] < qd[r - 1]) {
            float td = qd[r]; qd[r] = qd[r - 1]; qd[r - 1] = td;
            int   ti = qi[r]; qi[r] = qi[r - 1]; qi[r - 1] = ti;
          }
        }
      }
    }
  }
#pragma unroll
  for (int r = 0; r < kK; ++r) nbr[gq * kK + r] = qi[r];
}

// ------------------------------------------------------------- GEMM 1 ----
// h1[m, n] = sum_k X[m,k] * W1[n,k];  X = [rel_pos(3) | nbr_feat(64) | pad]
__global__ void gemm1_kernel(const float* __restrict__ feat, const float* __restrict__ pos,
                             const int* __restrict__ nbr, const float* __restrict__ W1,
                             float* __restrict__ h, float* __restrict__ part) {
  __shared__ __align__(16) float Xs[kRowsPerWG * kCinP];
  __shared__ __align__(16) float Ws[kCout * kCinP];
  __shared__ float redS[8][2][64];
  __shared__ float redQ[8][2][64];

  const int tid = threadIdx.x, wg = blockIdx.x;
  const int m0 = wg * kRowsPerWG;
  const int lane = tid & 31, wav = tid >> 5;
  const int lh = lane >> 4, lr = lane & 15;
  const int rg = wav & 3, cg = wav >> 2;

  v8f acc[4];
#pragma unroll
  for (int nt = 0; nt < 4; ++nt) acc[nt] = (v8f)0.0f;

  for (int kb = 0; kb < kCinP; kb += 64) {
    const int kc = (kCinP - kb < 64) ? (kCinP - kb) : 64;
    const int nseg = kc >> 2;
    __syncthreads();
    // stage X tile (gather + concat + pad)
    for (int idx = tid; idx < kRowsPerWG * nseg; idx += 256) {
      const int row = idx / nseg, seg = idx - row * nseg;
      const int m = m0 + row;
      const int bb = m >> 16;          // / (N*K)
      const int rem = m & 65535;
      const int ii = rem >> 4;         // dst point
      const int nid = nbr[m];          // src point
      const float* fr = feat + ((size_t)(bb * kN) + nid) * 64;
#pragma unroll
      for (int e = 0; e < 4; ++e) {
        const int col = kb + seg * 4 + e;
        float v;
        if (col < 3)        v = pos[((size_t)(bb * kN) + nid) * 3 + col] -
                                pos[((size_t)(bb * kN) + ii) * 3 + col];
        else if (col < kCin) v = fr[col - 3];
        else                 v = 0.0f;
        Xs[row * kCinP + seg * 4 + e] = v;
      }
    }
    // stage W1 (row-major [cout, cin], pad col 67)
    for (int idx = tid; idx < kCout * nseg; idx += 256) {
      const int n = idx / nseg, seg = idx - n * nseg;
#pragma unroll
      for (int e = 0; e < 4; ++e) {
        const int col = kb + seg * 4 + e;
        Ws[n * kCinP + seg * 4 + e] = (col < kCin) ? W1[(size_t)n * kCin + col] : 0.0f;
      }
    }
    __syncthreads();
    for (int ks = 0; ks < nseg; ++ks) {
      const int k0 = ks * 4 + 2 * lh;
      const v2f a = *(const v2f*)&Xs[(rg * 16 + lr) * kCinP + k0];
#pragma unroll
      for (int nt = 0; nt < 4; ++nt) {
        const v2f bf = *(const v2f*)&Ws[(cg * 64 + nt * 16 + lr) * kCinP + k0];
        acc[nt] = __builtin_amdgcn_wmma_f32_16x16x4_f32(false, a, false, bf,
                                                        (short)0, acc[nt], false, false);
      }
    }
  }
  // write h1 + per-channel partial sum / sumsq (deterministic)
#pragma unroll
  for (int nt = 0; nt < 4; ++nt) {
    float s = 0.0f, q = 0.0f;
#pragma unroll
    for (int r = 0; r < 8; ++r) {
      const float v = acc[nt][r];
      const int mrow = m0 + rg * 16 + lh * 8 + r;
      h[(size_t)mrow * kCout + cg * 64 + nt * 16 + lr] = v;
      s += v; q += v * v;
    }
    redS[wav][lh][nt * 16 + lr] = s;
    redQ[wav][lh][nt * 16 + lr] = q;
  }
  __syncthreads();
  if (tid < kCout) {
    const int cgr = tid >> 6, lc = tid & 63;
    float S = 0.0f, Q = 0.0f;
#pragma unroll
    for (int r2 = 0; r2 < 4; ++r2)
#pragma unroll
      for (int h2 = 0; h2 < 2; ++h2) {
        S += redS[cgr * 4 + r2][h2][lc];
        Q += redQ[cgr * 4 + r2][h2][lc];
      }
    part[(size_t)wg * 256 + tid] = S;
    part[(size_t)wg * 256 + 128 + tid] = Q;
  }
}

// --------------------------------------------------------- BN stats ------
__global__ void stats_kernel(const float* __restrict__ part, int nwg, float Minv,
                             const float* __restrict__ gamma, const float* __restrict__ beta,
                             float* __restrict__ sOut, float* __restrict__ tOut) {
  const int c = threadIdx.x;
  float S = 0.0f, Q = 0.0f;
  for (int w = 0; w < nwg; ++w) {           // fixed order -> deterministic
    S += part[(size_t)w * 256 + c];
    Q += part[(size_t)w * 256 + 128 + c];
  }
  const float mu = S * Minv;
  const float var = Q * Minv - mu * mu;
  const float sc = gamma[c] / sqrtf(var + 1e-5f);
  sOut[c] = sc;
  tOut[c] = beta[c] - mu * sc;              // bias cancels inside BN exactly
}

// ------------------------------------------------------------- GEMM 2 ----
// in-place: h2[m, n] = sum_k relu(s1[k]*h1[m,k]+t1[k]) * W2[n,k]
__global__ void gemm2_kernel(float* __restrict__ h, const float* __restrict__ W2,
                             const float* __restrict__ s1, const float* __restrict__ t1,
                             float* __restrict__ part) {
  __shared__ __align__(16) float Xs[kRowsPerWG * kCinP];
  __shared__ __align__(16) float Ws[kCout * kCinP];
  __shared__ float redS[8][2][64];
  __shared__ float redQ[8][2][64];
  __shared__ float sh_s[kCout], sh_t[kCout];

  const int tid = threadIdx.x, wg = blockIdx.x;
  const int m0 = wg * kRowsPerWG;
  const int lane = tid & 31, wav = tid >> 5;
  const int lh = lane >> 4, lr = lane & 15;
  const int rg = wav & 3, cg = wav >> 2;

  if (tid < kCout) { sh_s[tid] = s1[tid]; sh_t[tid] = t1[tid]; }

  v8f acc[4];
#pragma unroll
  for (int nt = 0; nt < 4; ++nt) acc[nt] = (v8f)0.0f;

  for (int kb = 0; kb < kCout; kb += 64) {
    __syncthreads();
    for (int idx = tid; idx < kRowsPerWG * 64; idx += 256) {
      const int r = idx >> 6, c = idx & 63;
      const int col = kb + c;
      float v = h[(size_t)(m0 + r) * kCout + col];
      v = fmaxf(v * sh_s[col] + sh_t[col], 0.0f);   // fused BN1 + ReLU
      Xs[r * kCinP + c] = v;
    }
    for (int idx = tid; idx < kCout * 64; idx += 256) {
      const int n = idx >> 6, c = idx & 63;
      Ws[n * kCinP + c] = W2[(size_t)n * kCout + kb + c];
    }
    __syncthreads();
    for (int ks = 0; ks < 16; ++ks) {
      const int k0 = ks * 4 + 2 * lh;
      const v2f a = *(const v2f*)&Xs[(rg * 16 + lr) * kCinP + k0];
#pragma unroll
      for (int nt = 0; nt < 4; ++nt) {
        const v2f bf = *(const v2f*)&Ws[(cg * 64 + nt * 16 + lr) * kCinP + k0];
        acc[nt] = __builtin_amdgcn_wmma_f32_16x16x4_f32(false, a, false, bf,
                                                        (short)0, acc[nt], false, false);
      }
    }
  }
#pragma unroll
  for (int nt = 0; nt < 4; ++nt) {
    float s = 0.0f, q = 0.0f;
#pragma unroll
    for (int r = 0; r < 8; ++r) {
      const float v = acc[nt][r];
      const int mrow = m0 + rg * 16 + lh * 8 + r;
      h[(size_t)mrow * kCout + cg * 64 + nt * 16 + lr] = v;   // safe in-place (own rows)
      s += v; q += v * v;
    }
    redS[wav][lh][nt * 16 + lr] = s;
    redQ[wav][lh][nt * 16 + lr] = q;
  }
  __syncthreads();
  if (tid < kCout) {
    const int cgr = tid >> 6, lc = tid & 63;
    float S = 0.0f, Q = 0.0f;
#pragma unroll
    for (int r2 = 0; r2 < 4; ++r2)
#pragma unroll
      for (int h2 = 0; h2 < 2; ++h2) {
        S += redS[cgr * 4 + r2][h2][lc];
        Q += redQ[cgr * 4 + r2][h2][lc];
      }
    part[(size_t)wg * 256 + tid] = S;
    part[(size_t)wg * 256 + 128 + tid] = Q;
  }
}

// ------------------------------------------------- BN2 + ReLU + max + gather
__global__ void finalize_kernel(const float* __restrict__ h, const float* __restrict__ pos,
                                const int* __restrict__ cent, const float* __restrict__ s2,
                                const float* __restrict__ t2, float* __restrict__ out) {
  __shared__ int s_row;
  const int bid = blockIdx.x, tid = threadIdx.x;
  const int b = bid >> 10, si = bid & 1023;
  if (tid == 0) s_row = cent[(size_t)b * kS + si];
  __syncthreads();
  const int i = s_row;
  if (tid < 3)
    out[((size_t)b * kS + si) * 3 + tid] = pos[((size_t)b * kN + i) * 3 + tid];
  const float sc = s2[tid], sh = t2[tid];
  const size_t base = (((size_t)b * kN + i) * kK) * kCout + tid;
  float m = 0.0f;   // ReLU outputs are >= 0
#pragma unroll
  for (int j = 0; j < kK; ++j) {
    const float v = fmaxf(h[base + (size_t)j * kCout] * sc + sh, 0.0f);
    m = fmaxf(m, v);
  }
  out[(size_t)kB * kS * 3 + ((size_t)b * kS + si) * kCout + tid] = m;
}

// ------------------------------------------------------------- launch ----
extern "C" void kernel_launch(void* const* d_in, const int* in_sizes, int n_in,
                              void* d_out, int out_size, void* d_ws, size_t ws_size,
                              hipStream_t stream) {
  (void)in_sizes; (void)n_in; (void)out_size; (void)ws_size;
  const float* feat = (const float*)d_in[0];
  const float* pos  = (const float*)d_in[1];
  const float* W1   = (const float*)d_in[2];
  // d_in[3] (b1) and d_in[7] (b2): biases cancel exactly inside BatchNorm
  const float* g1   = (const float*)d_in[4];
  const float* be1  = (const float*)d_in[5];
  const float* W2   = (const float*)d_in[6];
  const float* g2   = (const float*)d_in[8];
  const float* be2  = (const float*)d_in[9];

  char* ws = (char*)d_ws;
  int*   cent = (int*)(ws + OFF_CENT);
  int*   nbr  = (int*)(ws + OFF_KNN);
  float* s1   = (float*)(ws + OFF_S1);
  float* t1   = (float*)(ws + OFF_T1);
  float* s2   = (float*)(ws + OFF_S2);
  float* t2   = (float*)(ws + OFF_T2);
  float* part = (float*)(ws + OFF_PART);
  float* h    = (float*)(ws + OFF_H);
  const float Minv = 1.0f / (float)kM;

  fps_kernel<<<kB, 1024, 0, stream>>>(pos, cent);
  knn_kernel<<<(kB * kN) / 256, 256, 0, stream>>>(pos, nbr);
  gemm1_kernel<<<kNWG, 256, 0, stream>>>(feat, pos, nbr, W1, h, part);
  stats_kernel<<<1, kCout, 0, stream>>>(part, kNWG, Minv, g1, be1, s1, t1);
  gemm2_kernel<<<kNWG, 256, 0, stream>>>(h, W2, s1, t1, part);
  stats_kernel<<<1, kCout, 0, stream>>>(part, kNWG, Minv, g2, be2, s2, t2);
  finalize_kernel<<<kB * kS, kCout, 0, stream>>>(h, pos, cent, s2, t2, (float*)d_out);
}